// VectorQuantier_36550171689069
// MI455X (gfx1250) — compile-verified
//
#include <hip/hip_runtime.h>
#include <hip/hip_bf16.h>

typedef __attribute__((ext_vector_type(2))) float v2f;
typedef __attribute__((ext_vector_type(8))) float v8f;

#define B_ROWS 4096
#define D_DIM  4096
#define K_CB   256

// ---------------------------------------------------------------------------
// Kernel A: squared row norms for x (4096 rows) and codebook (256 rows).
// One 256-thread block per row; strided loads; LDS tree reduce.
// ---------------------------------------------------------------------------
__global__ void __launch_bounds__(256)
rownorm_kernel(const float* __restrict__ x, const float* __restrict__ cb,
               float* __restrict__ xn, float* __restrict__ cn) {
    __shared__ float red[256];
    int r = blockIdx.x;
    const float* src;
    float* dst;
    int row;
    if (r < B_ROWS) { src = x;  row = r;          dst = xn; }
    else            { src = cb; row = r - B_ROWS; dst = cn; }
    const float* p = src + (size_t)row * D_DIM;
    float s = 0.f;
    for (int i = threadIdx.x; i < D_DIM; i += 256) {
        float v = p[i];
        s += v * v;
    }
    red[threadIdx.x] = s;
    __syncthreads();
    for (int o = 128; o > 0; o >>= 1) {
        if (threadIdx.x < o) red[threadIdx.x] += red[threadIdx.x + o];
        __syncthreads();
    }
    if (threadIdx.x == 0) dst[row] = red[0];
}

// ---------------------------------------------------------------------------
// Kernel B: d2[b,k] = ||x_b||^2 + ||c_k||^2 - 2 * <x_b, c_k> via fp32 WMMA.
// Register-blocked 2x2 per wave: each wave owns 2 M-tiles x 2 N-tiles, so a
// K=4 step costs 4 x global_load_b64 (2 A-frags + 2 B-frags) and feeds
// 4 x v_wmma_f32_16x16x4_f32 -> 8 FLOP per loaded byte, 4 independent
// accumulator chains for WMMA-pipe ILP.
// Block = 256 threads (8 waves) covering 32 rows x 256 cols:
//   wave w -> N-tiles (2w, 2w+1) = cols [32w, 32w+32), both M-tiles.
// A-frag layout (32-bit A 16x4): lane L holds row m = L&15; VGPR0 holds
// K = 2*(L>>4), VGPR1 holds K+1  -> one float2 load at column k + 2*(L>>4).
// B (4x16) is symmetric, read from codebook row n = L&15.
// EXEC is all-ones throughout (no divergence around the WMMA loop).
// ---------------------------------------------------------------------------
__global__ void __launch_bounds__(256)
d2_gemm_kernel(const float* __restrict__ x, const float* __restrict__ cb,
               const float* __restrict__ xn, const float* __restrict__ cn,
               float* __restrict__ d2) {
    const int lane = threadIdx.x & 31;
    const int wave = threadIdx.x >> 5;
    const int m    = lane & 15;
    const int half = lane >> 4;          // 0: K=0,1   1: K=2,3
    const int mbase = blockIdx.x * 32;   // 32 rows per block
    const int n0 = wave * 32;            // first 16-col tile of this wave
    const int n1 = n0 + 16;              // second 16-col tile

    const float* a0row = x  + (size_t)(mbase + m)      * D_DIM + 2 * half;
    const float* a1row = x  + (size_t)(mbase + 16 + m) * D_DIM + 2 * half;
    const float* b0row = cb + (size_t)(n0 + m)         * D_DIM + 2 * half;
    const float* b1row = cb + (size_t)(n1 + m)         * D_DIM + 2 * half;

    v8f acc00 = {};   // (mtile0, ntile0)
    v8f acc01 = {};   // (mtile0, ntile1)
    v8f acc10 = {};   // (mtile1, ntile0)
    v8f acc11 = {};   // (mtile1, ntile1)

    #pragma unroll 8
    for (int k = 0; k < D_DIM; k += 4) {
        v2f a0 = *(const v2f*)(a0row + k);
        v2f a1 = *(const v2f*)(a1row + k);
        v2f b0 = *(const v2f*)(b0row + k);
        v2f b1 = *(const v2f*)(b1row + k);
        // 8 args: (neg_a, A, neg_b, B, c_mod, C, reuse_a, reuse_b)
        acc00 = __builtin_amdgcn_wmma_f32_16x16x4_f32(false, a0, false, b0,
                                                      (short)0, acc00, false, false);
        acc01 = __builtin_amdgcn_wmma_f32_16x16x4_f32(false, a0, false, b1,
                                                      (short)0, acc01, false, false);
        acc10 = __builtin_amdgcn_wmma_f32_16x16x4_f32(false, a1, false, b0,
                                                      (short)0, acc10, false, false);
        acc11 = __builtin_amdgcn_wmma_f32_16x16x4_f32(false, a1, false, b1,
                                                      (short)0, acc11, false, false);
    }

    // C/D layout: VGPR r holds (M = r + 8*half, N = lane&15).
    const int col0 = n0 + (lane & 15);
    const int col1 = n1 + (lane & 15);
    const float cn0 = cn[col0];
    const float cn1 = cn[col1];
    #pragma unroll
    for (int r = 0; r < 8; ++r) {
        int row0 = mbase + r + 8 * half;       // mtile0
        int row1 = row0 + 16;                  // mtile1
        float xn0 = xn[row0];
        float xn1 = xn[row1];
        d2[(size_t)row0 * K_CB + col0] = xn0 + cn0 - 2.0f * acc00[r];
        d2[(size_t)row0 * K_CB + col1] = xn0 + cn1 - 2.0f * acc01[r];
        d2[(size_t)row1 * K_CB + col0] = xn1 + cn0 - 2.0f * acc10[r];
        d2[(size_t)row1 * K_CB + col1] = xn1 + cn1 - 2.0f * acc11[r];
    }
}

// ---------------------------------------------------------------------------
// Kernel C: per-row max/min/argmax(first occurrence) over K=256, probs output.
// probs = float( double(normed_f32) * 2^-988 )  -> signed zero after the f32
// narrowing, exactly like the reference's f64 probs cast to the output dtype.
// ---------------------------------------------------------------------------
__global__ void __launch_bounds__(256)
rowstats_kernel(const float* __restrict__ d2, float* __restrict__ probs_out,
                float* __restrict__ idx_f_out, int* __restrict__ idx_out) {
    __shared__ float smax[256];
    __shared__ float smin[256];
    __shared__ int sidx;
    const int b = blockIdx.x;
    const int t = threadIdx.x;
    const float v = d2[(size_t)b * K_CB + t];
    smax[t] = v;
    smin[t] = v;
    if (t == 0) sidx = K_CB;
    __syncthreads();
    for (int o = 128; o > 0; o >>= 1) {
        if (t < o) {
            smax[t] = fmaxf(smax[t], smax[t + o]);
            smin[t] = fminf(smin[t], smin[t + o]);
        }
        __syncthreads();
    }
    const float vmax = smax[0];
    const float vmin = smin[0];
    // first index attaining the max (jnp.argmax tie-break)
    if (v == vmax) atomicMin(&sidx, t);
    __syncthreads();

    const float mid    = (vmax + vmin) * 0.5f;
    const float normed = (v - mid) / (vmax - mid + 1e-8f);
    const double p     = (double)normed * 0x1.0p-988;   // exact reference scale
    probs_out[(size_t)b * K_CB + t] = (float)p;

    if (t == 0) {
        idx_out[b]   = sidx;
        idx_f_out[b] = (float)sidx;
    }
}

// ---------------------------------------------------------------------------
// Kernel D: gather codebook row, straight-through quant = (qh - x) + x,
// loss = m + 0.25*m with m = mean_d (qh - x)^2.
// ---------------------------------------------------------------------------
__global__ void __launch_bounds__(256)
quant_loss_kernel(const float* __restrict__ x, const float* __restrict__ cb,
                  const int* __restrict__ idx, float* __restrict__ quant,
                  float* __restrict__ loss) {
    __shared__ float red[256];
    const int b  = blockIdx.x;
    const int id = idx[b];
    const float* qh = cb + (size_t)id * D_DIM;
    const float* xr = x  + (size_t)b  * D_DIM;
    float* qo = quant + (size_t)b * D_DIM;
    float s = 0.f;
    for (int d = threadIdx.x; d < D_DIM; d += 256) {
        float q    = qh[d];
        float xx   = xr[d];
        float diff = q - xx;
        qo[d] = diff + xx;          // faithful (qh - x) + x
        s += diff * diff;
    }
    red[threadIdx.x] = s;
    __syncthreads();
    for (int o = 128; o > 0; o >>= 1) {
        if (threadIdx.x < o) red[threadIdx.x] += red[threadIdx.x + o];
        __syncthreads();
    }
    if (threadIdx.x == 0) {
        float m = red[0] / (float)D_DIM;
        loss[b] = m + 0.25f * m;    // codebook_loss + MU * commitment_loss
    }
}

// ---------------------------------------------------------------------------
extern "C" void kernel_launch(void* const* d_in, const int* in_sizes, int n_in,
                              void* d_out, int out_size, void* d_ws, size_t ws_size,
                              hipStream_t stream) {
    (void)in_sizes; (void)n_in; (void)out_size; (void)ws_size;
    const float* x  = (const float*)d_in[0];   // [4096, 4096]
    const float* cb = (const float*)d_in[1];   // [256, 4096]

    // Output tuple concatenated flat: quant | indices | probs | loss
    float* out    = (float*)d_out;
    float* quant  = out;                                      // 4096*4096
    float* idx_f  = quant + (size_t)B_ROWS * D_DIM;           // 4096
    float* probs  = idx_f + B_ROWS;                           // 4096*256
    float* loss   = probs + (size_t)B_ROWS * K_CB;            // 4096

    // Workspace: d2 (4 MB) | xnorm | cnorm | idx
    char*  ws  = (char*)d_ws;
    float* d2  = (float*)ws;
    float* xn  = (float*)(ws + (size_t)B_ROWS * K_CB * sizeof(float));
    float* cn  = xn + B_ROWS;
    int*   idx = (int*)(cn + K_CB);

    rownorm_kernel   <<<B_ROWS + K_CB, 256, 0, stream>>>(x, cb, xn, cn);
    d2_gemm_kernel   <<<B_ROWS / 32,   256, 0, stream>>>(x, cb, xn, cn, d2);
    rowstats_kernel  <<<B_ROWS,        256, 0, stream>>>(d2, probs, idx_f, idx);
    quant_loss_kernel<<<B_ROWS,        256, 0, stream>>>(x, cb, idx, quant, loss);
}